// MultiHeadedAttention_3_18631568130098
// MI455X (gfx1250) — compile-verified
//
#include <hip/hip_runtime.h>

typedef __attribute__((ext_vector_type(2))) float v2f;
typedef __attribute__((ext_vector_type(8))) float v8f;

#define Bsz 2
#define Nn  512
#define Cc  64
#define Hh  8

__global__ __launch_bounds__(256) void mha_roi_softmax_kernel(
    const float* __restrict__ q, const float* __restrict__ k,
    const float* __restrict__ roi, const float* __restrict__ W,
    const float* __restrict__ bias, float* __restrict__ out)
{
    __shared__ float attnS[Nn * Hh];   // 16 KB: attn[n][h] for this (b,m) row
    __shared__ float roiS[Nn];         // 2 KB
    __shared__ float invS[Hh];

    const int row   = blockIdx.x;      // flat (b*N + m)
    const int tid   = threadIdx.x;
    const int wave  = tid >> 5;
    const int lane  = tid & 31;
    const int hlane = lane & 15;       // N index of WMMA tile = head (padded to 16)
    const int hi    = lane >> 4;       // half-wave selector

    // --- stage roi row into LDS (coalesced) ---
    const float* roiRow = roi + (size_t)row * Nn;
    roiS[tid]       = roiRow[tid];
    roiS[tid + 256] = roiRow[tid + 256];

    // --- build B fragments (weights) in registers, once per wave ---
    // B 4x16 layout: lane L holds col N=L%16; VGPR v holds K = v + 2*(L/16).
    // Wq[h][c] = W[h*2C + c], Wk[h][c] = W[h*2C + 64 + c]; heads >= 8 padded with 0.
    v2f bq[16], bk[16];
#pragma unroll
    for (int i = 0; i < 16; ++i) {
        if (hlane < Hh) {
            const float* wr = W + hlane * (2 * Cc) + 4 * i + 2 * hi;
            bq[i].x = wr[0];      bq[i].y = wr[1];
            bk[i].x = wr[Cc];     bk[i].y = wr[Cc + 1];
        } else {
            bq[i].x = 0.f; bq[i].y = 0.f;
            bk[i].x = 0.f; bk[i].y = 0.f;
        }
    }
    const float bv = (hlane < Hh) ? bias[hlane] : 0.f;

    const float* qRow = q + (size_t)row * (Nn * Cc);
    const float* kRow = k + (size_t)row * (Nn * Cc);

    __syncthreads();   // roiS ready

    // --- phase 1: projection via fp32 WMMA, exp*mask into LDS ---
    // 32 tiles of 16 positions; 8 waves -> 4 tiles each.
    for (int t = wave; t < Nn / 16; t += 8) {
        const int n0 = t * 16;
        // A 16x4 layout: lane L holds row M=L%16 (position), VGPR v: K=v+2*(L/16)
        const float* qp = qRow + (size_t)(n0 + hlane) * Cc + 2 * hi;
        const float* kp = kRow + (size_t)(n0 + hlane) * Cc + 2 * hi;

        v8f c = { bv, bv, bv, bv, bv, bv, bv, bv };  // bias folded into accumulator
#pragma unroll
        for (int i = 0; i < 16; ++i) {
            v2f a = *(const v2f*)(qp + 4 * i);
            c = __builtin_amdgcn_wmma_f32_16x16x4_f32(
                    false, a, false, bq[i], (short)0, c, false, false);
        }
#pragma unroll
        for (int i = 0; i < 16; ++i) {
            v2f a = *(const v2f*)(kp + 4 * i);
            c = __builtin_amdgcn_wmma_f32_16x16x4_f32(
                    false, a, false, bk[i], (short)0, c, false, false);
        }

        // C/D layout: lane L, VGPR v -> position p = v + 8*(L/16), head = L%16
        if (hlane < Hh) {
#pragma unroll
            for (int v = 0; v < 8; ++v) {
                const int p = n0 + v + 8 * hi;
                attnS[p * Hh + hlane] = __expf(c[v]) * roiS[p];
            }
        }
    }

    __syncthreads();

    // --- phase 2: per-head row sums; wave w owns head w ---
    if (wave < Hh) {
        float s = 0.f;
        for (int n = lane; n < Nn; n += 32) s += attnS[n * Hh + wave];
#pragma unroll
        for (int off = 16; off > 0; off >>= 1) s += __shfl_xor(s, off, 32);
        if (lane == 0) invS[wave] = 1.0f / s;
    }

    __syncthreads();

    // --- phase 3: normalize + coalesced float4 stores (16 KB per block) ---
    const float4 inv0 = make_float4(invS[0], invS[1], invS[2], invS[3]);
    const float4 inv1 = make_float4(invS[4], invS[5], invS[6], invS[7]);
    const float4* a4  = (const float4*)attnS;
    float4* out4 = (float4*)out + (size_t)row * (Nn * Hh / 4);
#pragma unroll
    for (int i = 0; i < 4; ++i) {
        const int kk = tid + i * 256;          // 1024 float4 chunks total
        float4 av = a4[kk];
        const float4 iv = (kk & 1) ? inv1 : inv0;
        av.x *= iv.x; av.y *= iv.y; av.z *= iv.z; av.w *= iv.w;
        out4[kk] = av;
    }
}

extern "C" void kernel_launch(void* const* d_in, const int* in_sizes, int n_in,
                              void* d_out, int out_size, void* d_ws, size_t ws_size,
                              hipStream_t stream) {
    const float* q    = (const float*)d_in[0];
    const float* k    = (const float*)d_in[1];
    const float* roi  = (const float*)d_in[2];
    const float* W    = (const float*)d_in[3];
    const float* bias = (const float*)d_in[4];
    float* out = (float*)d_out;

    dim3 grid(Bsz * Nn);   // 1024 workgroups, one per (b, m) row
    dim3 block(256);       // 8 wave32
    mha_roi_softmax_kernel<<<grid, block, 0, stream>>>(q, k, roi, W, bias, out);
}